// Recognizer_GMM_12945031430862
// MI455X (gfx1250) — compile-verified
//
#include <hip/hip_runtime.h>
#include <math.h>

#define CH    64
#define EPSG  1e-15f
#define NB    8
#define EDGES 320000
#define NMAX  40000
#define RSPLIT 32   // rank-kernel blocks per graph

typedef __attribute__((ext_vector_type(2))) float v2f;
typedef __attribute__((ext_vector_type(8))) float v8f;

// ---------------------------------------------------------------- init kernels
__global__ void k_init_edges(const int* __restrict__ ei, int* __restrict__ srcB,
                             int* __restrict__ dstB, float* __restrict__ emask, int E) {
    int e = blockIdx.x * blockDim.x + threadIdx.x;
    if (e < E) { srcB[e] = ei[e]; dstB[e] = ei[E + e]; emask[e] = 1.0f; }
}

__global__ void k_init_x0(const float* __restrict__ x, const float* __restrict__ f,
                          float* __restrict__ xin, int N) {
    int i = blockIdx.x * blockDim.x + threadIdx.x;
    if (i < N * CH) {
        int n = i >> 6, c = i & 63;
        float v = 0.f;
        if (c < 3)      v = x[n * 3 + c];
        else if (c < 8) v = f[n * 5 + (c - 3)];
        xin[i] = v;
    }
}

__global__ void k_zero(float* __restrict__ p, int n) {
    int i = blockIdx.x * blockDim.x + threadIdx.x;
    if (i < n) p[i] = 0.f;
}

// Pack W = [g | root] into [64 x 256], zero-padding rows >= cin (layer 0: cin=8)
__global__ void k_pack_w(const float* __restrict__ g, const float* __restrict__ root,
                         float* __restrict__ W, int cin) {
    int i = blockIdx.x * blockDim.x + threadIdx.x;
    if (i < 64 * 256) {
        int r = i >> 8, c = i & 255;
        float v = 0.f;
        if (r < cin) v = (c < 192) ? g[r * 192 + c] : root[r * 64 + (c - 192)];
        W[i] = v;
    }
}

// ---------------------------------------------------------------- WMMA GEMM
// Y[rows,256] = X[rows,64] @ W[64,256], fp32, one 16x16 tile per wave.
// 16 waves/block cover one 16-row stripe across all 16 column tiles.
__global__ void k_gemm(const float* __restrict__ X, const float* __restrict__ W,
                       float* __restrict__ Y) {
    int wid  = threadIdx.x >> 5;      // column tile 0..15
    int lane = threadIdx.x & 31;
    int m    = lane & 15;             // A row / B col / D col within tile
    int half = lane >> 4;             // K-pair select (A/B), M+8 select (D)
    int row0 = blockIdx.x * 16;
    int col0 = wid * 16;

    v8f acc = {};
    const float* xrow = X + (size_t)(row0 + m) * 64;
#pragma unroll
    for (int kk = 0; kk < 16; ++kk) {
        int kb = kk * 4 + half * 2;
        v2f a = *(const v2f*)(xrow + kb);                 // A: rows x K, 2 K-vals/lane
        v2f b;
        b.x = W[(size_t)kb * 256 + col0 + m];             // B: K x cols
        b.y = W[(size_t)(kb + 1) * 256 + col0 + m];
        acc = __builtin_amdgcn_wmma_f32_16x16x4_f32(
            /*neg_a=*/false, a, /*neg_b=*/false, b,
            /*c_mod=*/(short)0, acc, /*reuse_a=*/false, /*reuse_b=*/false);
    }
#pragma unroll
    for (int r = 0; r < 8; ++r)
        Y[(size_t)(row0 + r + half * 8) * 256 + col0 + m] = acc[r];
}

// ---------------------------------------------------------------- edge messages
// Precompute gaussian kernel weights once per edge (3 exps) instead of 64x.
__global__ void k_gauss(const float* __restrict__ eattr, const float* __restrict__ mu,
                        const float* __restrict__ sigma, float* __restrict__ gauss, int E) {
    int e = blockIdx.x * blockDim.x + threadIdx.x;
    if (e >= E) return;
    float ea = eattr[e];
#pragma unroll
    for (int k = 0; k < 3; ++k) {
        float df = ea - mu[k];
        gauss[e * 3 + k] = expf(-0.5f * df * df / (EPSG + sigma[k] * sigma[k]));
    }
}

// One 64-thread group per edge; gauss weights broadcast from cache.
__global__ void k_edges(const int* __restrict__ srcB, const int* __restrict__ dstB,
                        const float* __restrict__ emask, const float* __restrict__ gauss,
                        const float* __restrict__ h, float* __restrict__ agg,
                        float* __restrict__ deg, int E) {
    int t = blockIdx.x * blockDim.x + threadIdx.x;
    int e = t >> 6, c = t & 63;
    if (e >= E) return;
    float m = emask[e];
    if (m <= 0.f) return;
    int s = srcB[e], d = dstB[e];
    float g0 = gauss[e * 3 + 0], g1 = gauss[e * 3 + 1], g2 = gauss[e * 3 + 2];
    float msg = g0 * h[(size_t)s * 256 + c]
              + g1 * h[(size_t)s * 256 + 64 + c]
              + g2 * h[(size_t)s * 256 + 128 + c];
    atomicAdd(&agg[(size_t)d * 64 + c], msg);
    if (c == 0) atomicAdd(&deg[d], 1.0f);
}

// out = relu(agg/max(deg,1) + x@root + bias)
__global__ void k_finalize(const float* __restrict__ agg, const float* __restrict__ deg,
                           const float* __restrict__ hout, const float* __restrict__ bias,
                           float* __restrict__ hact, int N) {
    int i = blockIdx.x * blockDim.x + threadIdx.x;
    if (i < N * CH) {
        int n = i >> 6, c = i & 63;
        float dg = deg[n]; dg = dg > 1.f ? dg : 1.f;
        float v = agg[i] / dg + hout[(size_t)n * 256 + 192 + c] + bias[c];
        hact[i] = v > 0.f ? v : 0.f;
    }
}

// ---------------------------------------------------------------- TopK pooling
// Stage 1: scores to global buffer (thread per node).
__global__ void k_score(const float* __restrict__ hact, const float* __restrict__ w,
                        float* __restrict__ sc, int Ntot) {
    int n = blockIdx.x * blockDim.x + threadIdx.x;
    if (n >= Ntot) return;
    float nrm = 0.f;
#pragma unroll
    for (int c = 0; c < CH; ++c) nrm += w[c] * w[c];
    float inv = 1.0f / sqrtf(nrm);
    float dot = 0.f;
    for (int c = 0; c < CH; ++c) dot += hact[(size_t)n * 64 + c] * w[c];
    sc[n] = tanhf(dot * inv);
}

// Stage 2: exact rank (desc score, lower-index tie-break) == jax.lax.top_k order.
// grid = (NB, RSPLIT); each block ranks a slice of one graph's nodes, streaming
// the graph's full score vector through a 4KB LDS tile. Chip-wide occupancy.
__global__ void k_rank(const float* __restrict__ sc, const float* __restrict__ hact,
                       float* __restrict__ xnew, int* __restrict__ map,
                       int n_cur, int kkeep) {
    __shared__ float tile[1024];
    int g     = blockIdx.x;
    int chunk = (n_cur + RSPLIT - 1) / RSPLIT;           // <= 157 <= blockDim
    int i     = blockIdx.y * chunk + threadIdx.x;        // node within graph
    bool own  = (threadIdx.x < chunk) && (i < n_cur);
    const float* scg = sc + (size_t)g * n_cur;

    float si = own ? scg[i] : 0.f;
    int rank = 0;
    for (int base = 0; base < n_cur; base += 1024) {
        int lim = n_cur - base; if (lim > 1024) lim = 1024;
        for (int t = threadIdx.x; t < lim; t += blockDim.x) tile[t] = scg[base + t];
        __syncthreads();
        if (own) {
            for (int j = 0; j < lim; ++j) {
                float sj = tile[j];
                rank += (sj > si) || (sj == si && (base + j) < i);
            }
        }
        __syncthreads();
    }
    if (own) {
        size_t gid = (size_t)g * n_cur + i;
        if (rank < kkeep) {
            int ng = g * kkeep + rank;
            map[gid] = ng;
            for (int c = 0; c < CH; ++c)
                xnew[(size_t)ng * 64 + c] = hact[gid * 64 + c] * si;
        } else {
            map[gid] = -1;
        }
    }
}

__global__ void k_remap(const int* __restrict__ map, int* __restrict__ srcB,
                        int* __restrict__ dstB, float* __restrict__ emask, int E) {
    int e = blockIdx.x * blockDim.x + threadIdx.x;
    if (e >= E) return;
    float m = emask[e];
    int ns = map[srcB[e]], nd = map[dstB[e]];
    if (m > 0.f && ns >= 0 && nd >= 0) { srcB[e] = ns; dstB[e] = nd; }
    else { srcB[e] = 0; dstB[e] = 0; emask[e] = 0.f; }
}

// per-graph max|mean readout: 256 threads = 4 row-chunks x 64 channels, LDS reduce
__global__ void k_readout(const float* __restrict__ xn, float* __restrict__ reads, int kkeep) {
    __shared__ float smx[256], ssm[256];
    int g = blockIdx.x, c = threadIdx.x & 63, q = threadIdx.x >> 6;
    float mx = -3.402823e38f, sm = 0.f;
    for (int i = q; i < kkeep; i += 4) {
        float v = xn[((size_t)g * kkeep + i) * 64 + c];
        mx = fmaxf(mx, v); sm += v;
    }
    smx[threadIdx.x] = mx; ssm[threadIdx.x] = sm;
    __syncthreads();
    if (q == 0) {
#pragma unroll
        for (int r = 1; r < 4; ++r) {
            mx = fmaxf(mx, smx[r * 64 + c]);
            sm += ssm[r * 64 + c];
        }
        reads[g * 128 + c]      += mx;
        reads[g * 128 + 64 + c] += sm / (float)kkeep;
    }
}

// ---------------------------------------------------------------- MLP head
__global__ void k_mlp(const float* __restrict__ reads,
                      const float* __restrict__ W1, const float* __restrict__ b1,
                      const float* __restrict__ W2, const float* __restrict__ b2,
                      const float* __restrict__ W3, const float* __restrict__ b3,
                      float* __restrict__ out) {
    __shared__ float z1[NB * 64];
    __shared__ float z2[NB * 64];
    int b = threadIdx.x >> 6, c = threadIdx.x & 63;   // 512 threads = 8 x 64
    float z = 0.f;
    for (int i = 0; i < 128; ++i) z += reads[b * 128 + i] * W1[i * 64 + c];
    z += b1[c];
    z1[b * 64 + c] = z > 0.f ? z : 0.f;
    __syncthreads();
    z = 0.f;
    for (int i = 0; i < 64; ++i) z += z1[b * 64 + i] * W2[i * 64 + c];
    z += b2[c];
    z2[b * 64 + c] = z > 0.f ? z : 0.f;
    __syncthreads();
    if (c < 10) {
        z = 0.f;
        for (int i = 0; i < 64; ++i) z += z2[b * 64 + i] * W3[i * 10 + c];
        out[b * 10 + c] = z + b3[c];
    }
}

// ---------------------------------------------------------------- launch
extern "C" void kernel_launch(void* const* d_in, const int* in_sizes, int n_in,
                              void* d_out, int out_size, void* d_ws, size_t ws_size,
                              hipStream_t stream) {
    // pytree flatten order: x, fields, edge_attr, params{convs[4]{bias,g,mu,root,sigma},
    // lin1_b,lin1_w,lin2_b,lin2_w,lin3_b,lin3_w, pool_w[4]}, edge_index, batch
    const float* x      = (const float*)d_in[0];
    const float* fields = (const float*)d_in[1];
    const float* eattr  = (const float*)d_in[2];
    const float *cbias[4], *cg[4], *cmu[4], *croot[4], *csig[4];
    for (int i = 0; i < 4; ++i) {
        int base = 3 + 5 * i;
        cbias[i] = (const float*)d_in[base + 0];
        cg[i]    = (const float*)d_in[base + 1];
        cmu[i]   = (const float*)d_in[base + 2];
        croot[i] = (const float*)d_in[base + 3];
        csig[i]  = (const float*)d_in[base + 4];
    }
    const float* l1b = (const float*)d_in[23];
    const float* l1w = (const float*)d_in[24];
    const float* l2b = (const float*)d_in[25];
    const float* l2w = (const float*)d_in[26];
    const float* l3b = (const float*)d_in[27];
    const float* l3w = (const float*)d_in[28];
    const float* pw[4] = { (const float*)d_in[29], (const float*)d_in[30],
                           (const float*)d_in[31], (const float*)d_in[32] };
    const int* ei = (const int*)d_in[33];

    // workspace carving
    char* ws = (char*)d_ws;
    size_t off = 0;
    auto carve = [&](size_t bytes) { void* p = ws + off; off += (bytes + 255) & ~(size_t)255; return p; };
    float* xin   = (float*)carve((size_t)NMAX * 64 * 4);
    float* hout  = (float*)carve((size_t)NMAX * 256 * 4);
    float* hact  = (float*)carve((size_t)NMAX * 64 * 4);
    float* agg   = (float*)carve((size_t)NMAX * 64 * 4);
    float* deg   = (float*)carve((size_t)NMAX * 4);
    int*   map   = (int*)  carve((size_t)NMAX * 4);
    float* sc    = (float*)carve((size_t)NMAX * 4);
    int*   srcB  = (int*)  carve((size_t)EDGES * 4);
    int*   dstB  = (int*)  carve((size_t)EDGES * 4);
    float* emask = (float*)carve((size_t)EDGES * 4);
    float* gauss = (float*)carve((size_t)EDGES * 3 * 4);
    float* W     = (float*)carve((size_t)64 * 256 * 4);
    float* reads = (float*)carve((size_t)NB * 128 * 4);

    const int ns[5] = { 5000, 4000, 3200, 2560, 2048 };

    k_init_edges<<<(EDGES + 255) / 256, 256, 0, stream>>>(ei, srcB, dstB, emask, EDGES);
    k_init_x0<<<(NMAX * 64 + 255) / 256, 256, 0, stream>>>(x, fields, xin, NMAX);
    k_zero<<<(NB * 128 + 255) / 256, 256, 0, stream>>>(reads, NB * 128);

    for (int i = 0; i < 4; ++i) {
        int Ntot = NB * ns[i];
        int kk   = ns[i + 1];
        int cin  = (i == 0) ? 8 : 64;

        k_pack_w<<<(64 * 256 + 255) / 256, 256, 0, stream>>>(cg[i], croot[i], W, cin);
        k_zero<<<(Ntot * 64 + 255) / 256, 256, 0, stream>>>(agg, Ntot * 64);
        k_zero<<<(Ntot + 255) / 256, 256, 0, stream>>>(deg, Ntot);

        k_gemm<<<Ntot / 16, 512, 0, stream>>>(xin, W, hout);

        k_gauss<<<(EDGES + 255) / 256, 256, 0, stream>>>(eattr, cmu[i], csig[i], gauss, EDGES);
        k_edges<<<(EDGES * 64 + 255) / 256, 256, 0, stream>>>(
            srcB, dstB, emask, gauss, hout, agg, deg, EDGES);

        k_finalize<<<(Ntot * 64 + 255) / 256, 256, 0, stream>>>(
            agg, deg, hout, cbias[i], hact, Ntot);

        k_score<<<(Ntot + 255) / 256, 256, 0, stream>>>(hact, pw[i], sc, Ntot);
        k_rank<<<dim3(NB, RSPLIT), 256, 0, stream>>>(sc, hact, xin, map, ns[i], kk);
        k_remap<<<(EDGES + 255) / 256, 256, 0, stream>>>(map, srcB, dstB, emask, EDGES);
        k_readout<<<NB, 256, 0, stream>>>(xin, reads, kk);
    }

    k_mlp<<<1, 512, 0, stream>>>(reads, l1w, l1b, l2w, l2b, l3w, l3b, (float*)d_out);
}